// KinematicLayer_13718125543844
// MI455X (gfx1250) — compile-verified
//
#include <hip/hip_runtime.h>
#include <math.h>

// ---------------------------------------------------------------------------
// KinematicLayer rollout, MI455X (gfx1250, wave32). Memory-bound (~543 MB
// total -> ~23us @ 23.3 TB/s HBM). 64 threads/block own 64 consecutive
// trajectories; the block's speed/yaw inputs are contiguous 16KB regions,
// staged to LDS with global_load_async_to_lds_b128 (ASYNCcnt) into padded
// rows (17 float4 = 272B stride -> 16B-aligned chunks, bank-friendly).
// Single LDS-fed compute loop (tail blocks stage via guarded ds stores), and
// sincosf (one range reduction per step) keep the hot code compact.
// ---------------------------------------------------------------------------

#define T_STEPS        64
#define ROWS           64                      // threads per block (2 waves)
#define ROW_F4         17                      // float4 per padded row (64 floats + 16B pad)
#define ROW_BYTES      (ROW_F4 * 16)           // 272
#define LDS_ARRAY_B    (ROWS * ROW_BYTES)      // 17408 bytes per array
#define COPY_ITERS     16                      // 16B chunks per thread per array

__global__ __launch_bounds__(ROWS) void kin_rollout_kernel(
    const float* __restrict__ speed,        // (B, 64)
    const float* __restrict__ yaw,          // (B, 64)
    const float* __restrict__ init_pos,     // (B, 2)
    const float* __restrict__ init_heading, // (B,)
    float*       __restrict__ out,          // (B, 64, 2)
    int B)
{
    __shared__ __align__(16) float4 lds_sp[ROWS * ROW_F4];
    __shared__ __align__(16) float4 lds_yw[ROWS * ROW_F4];

    const int  tid  = (int)threadIdx.x;
    const int  b0   = (int)blockIdx.x * ROWS;
    const int  b    = b0 + tid;
    const bool full = (b0 + ROWS) <= B;

    const float* gsp = speed + (size_t)b0 * T_STEPS;  // contiguous 16KB per block
    const float* gyw = yaw   + (size_t)b0 * T_STEPS;

    if (full) {
        // Coalesced async staging: low 32 bits of a generic shared pointer
        // are the LDS byte offset.
        unsigned sp_base = (unsigned)(size_t)(&lds_sp[0]);
        unsigned yw_base = (unsigned)(size_t)(&lds_yw[0]);
        #pragma unroll
        for (int i = 0; i < COPY_ITERS; ++i) {
            int k   = i * ROWS + tid;          // 16B-chunk id, coalesced across lanes
            int row = k >> 4;
            int cin = k & 15;
            unsigned loff = (unsigned)(row * ROW_BYTES + cin * 16);
            unsigned lsp  = sp_base + loff;
            unsigned lyw  = yw_base + loff;
            int      gof  = k * 16;            // byte offset from block base
            asm volatile("global_load_async_to_lds_b128 %0, %1, %2"
                         :: "v"(lsp), "v"(gof), "s"(gsp) : "memory");
            asm volatile("global_load_async_to_lds_b128 %0, %1, %2"
                         :: "v"(lyw), "v"(gof), "s"(gyw) : "memory");
        }
        asm volatile("s_wait_asynccnt 0" ::: "memory");
    } else if (b < B) {
        // Ragged tail block (absent for B = 524288): stage own row via ds stores.
        const float4* g1 = (const float4*)(gsp + (size_t)tid * T_STEPS);
        const float4* g2 = (const float4*)(gyw + (size_t)tid * T_STEPS);
        #pragma unroll 1
        for (int q = 0; q < T_STEPS / 4; ++q) {
            lds_sp[tid * ROW_F4 + q] = g1[q];
            lds_yw[tid * ROW_F4 + q] = g2[q];
        }
    }

    __syncthreads();   // publish LDS tiles (each wave drained its own ASYNCcnt)

    if (b < B) {
        const float DIAG = 2202.9071700822983f;   // sqrt(1920^2 + 1080^2)
        const float DT   = 0.1f;

        float2 ip = ((const float2*)init_pos)[b];
        float x   = ip.x * 1920.0f;
        float y   = ip.y * 1080.0f;
        float psi = init_heading[b];
        float s, c;
        sincosf(psi, &s, &c);
        float dec = 1.0f;                         // 0.97^t, t = 0

        float4* orow4 = (float4*)(out + (size_t)b * T_STEPS * 2);
        const int base = tid * ROW_F4;            // direct __shared__ indexing -> ds_load_b128

        #pragma unroll 1
        for (int q = 0; q < T_STEPS / 4; ++q) {
            float4 v4 = lds_sp[base + q];
            float4 w4 = lds_yw[base + q];
            float vs[4] = {v4.x, v4.y, v4.z, v4.w};
            float ws[4] = {w4.x, w4.y, w4.z, w4.w};
            float ox[4], oy[4];

            #pragma unroll
            for (int j = 0; j < 4; ++j) {
                float v = vs[j] * DIAG;
                float w = ws[j] * dec;
                dec *= 0.97f;

                float is_s = (fabsf(w) < 0.01f) ? 1.0f : 0.0f;
                float psin = psi + w * DT;
                float sn, cn;
                sincosf(psin, &sn, &cn);

                float wsafe  = w + is_s * 0.0001f;
                float radius = v / wsafe;

                float dx = is_s * (v * c * DT) + (1.0f - is_s) * (radius * (sn - s));
                float dy = is_s * (v * s * DT) + (1.0f - is_s) * (-radius * (cn - c));
                x += dx;
                y += dy;

                ox[j] = x / 1920.0f;
                oy[j] = y / 1080.0f;

                psi = psin; s = sn; c = cn;
            }

            float4 o0; o0.x = ox[0]; o0.y = oy[0]; o0.z = ox[1]; o0.w = oy[1];
            float4 o1; o1.x = ox[2]; o1.y = oy[2]; o1.z = ox[3]; o1.w = oy[3];
            orow4[2 * q + 0] = o0;
            orow4[2 * q + 1] = o1;
        }
    }
}

extern "C" void kernel_launch(void* const* d_in, const int* in_sizes, int n_in,
                              void* d_out, int out_size, void* d_ws, size_t ws_size,
                              hipStream_t stream) {
    const float* speed        = (const float*)d_in[0];
    const float* yaw_rate     = (const float*)d_in[1];
    const float* init_pos     = (const float*)d_in[2];
    const float* init_heading = (const float*)d_in[3];
    float*       out          = (float*)d_out;

    const int B = in_sizes[3];                 // init_heading is (B,)
    const int blocks = (B + ROWS - 1) / ROWS;  // B = 524288 -> 8192 blocks, no tail

    kin_rollout_kernel<<<blocks, ROWS, 0, stream>>>(
        speed, yaw_rate, init_pos, init_heading, out, B);
}